// VolumeRenderer_6012954214893
// MI455X (gfx1250) — compile-verified
//
#include <hip/hip_runtime.h>
#include <cstdint>

// ---------------------------------------------------------------------------
// Volume rendering (NeRF-style weights + weighted feature/depth reduction)
// MI455X / gfx1250: bandwidth-bound streaming scan.
// Data path: global -> LDS via CDNA5 async ops (global_load_async_to_lds_b128,
// s_wait_asynccnt), TRIPLE-buffered (1 barrier/chunk), addresses strength-
// reduced to pure increments, non-temporal hints on the one-touch stream.
// ---------------------------------------------------------------------------

#define NSAMP 128                 // samples per ray (reference N)
#define RPB   128                 // rays per block == threads per block
#define CS    8                   // samples per staged chunk
#define NCHUNK (NSAMP / CS)       // 16
#define NBUF  3                   // triple buffering
#define DEN_PITCH 12              // floats/row: 8 data + 4 pad (16B-aligned rows)
#define FEAT_PITCH 28             // floats/row: 24 data + 4 pad (16B-aligned rows)
#define BUF_FLOATS (RPB * DEN_PITCH * 2 + RPB * FEAT_PITCH)  // 6656
#define BUF_BYTES  (BUF_FLOATS * 4)                          // 26624
#define FAR_DELTA 1.0e10f
#define NQ 10                     // async issues per thread per chunk (2+2+6)

__device__ __forceinline__ uint32_t lds_byte_off(const void* p) {
  return (uint32_t)(uintptr_t)p;  // LDS-aperture flat address: low 32 bits
}

__device__ __forceinline__ void async_ld16(uint32_t lds_off, uint64_t gaddr) {
  // CDNA5 async copy: global -> LDS at the vdst offset, 16B, tracked by ASYNCcnt.
  asm volatile("global_load_async_to_lds_b128 %0, %1, off th:TH_LOAD_NT"
               :: "v"(lds_off), "v"(gaddr)
               : "memory");
}

#define WAIT_ASYNCCNT(n) asm volatile("s_wait_asynccnt %0" :: "i"(n) : "memory")

__global__ __launch_bounds__(RPB) void volrender_kernel(
    const float* __restrict__ dens, const float* __restrict__ feat,
    const float* __restrict__ depth, float* __restrict__ out, int Btot) {
  __shared__ __align__(16) float lds[NBUF * BUF_FLOATS];

  const int t = threadIdx.x;                 // local ray index
  const int rayBase = blockIdx.x * RPB;
  const uint32_t ldsBase = lds_byte_off(lds);

  // ---- one-time setup of the 10 copy slots (global addr + LDS offset) ----
  uint64_t gA[NQ];     // global byte addresses, auto-incremented per chunk
  uint32_t lO[NQ];     // LDS byte offsets within buffer 0 (ldsBase folded in)

#pragma unroll
  for (int j = 0; j < 2; ++j) {              // density + depth: 2x16B per ray
    const int k = j * RPB + t;
    const int r = k >> 1, blk = k & 1;
    const int ray = min(rayBase + r, Btot - 1);
    gA[j]     = (uint64_t)(uintptr_t)((const char*)dens  + (size_t)ray * (NSAMP * 4) + blk * 16);
    gA[2 + j] = (uint64_t)(uintptr_t)((const char*)depth + (size_t)ray * (NSAMP * 4) + blk * 16);
    lO[j]     = ldsBase + (uint32_t)(r * (DEN_PITCH * 4) + blk * 16);
    lO[2 + j] = ldsBase + (uint32_t)(RPB * DEN_PITCH * 4 + r * (DEN_PITCH * 4) + blk * 16);
  }
#pragma unroll
  for (int j = 0; j < 6; ++j) {              // feature: 6x16B per ray
    const int k = j * RPB + t;
    const int r = k / 6, blk = k % 6;
    const int ray = min(rayBase + r, Btot - 1);
    gA[4 + j] = (uint64_t)(uintptr_t)((const char*)feat + (size_t)ray * (NSAMP * 12) + blk * 16);
    lO[4 + j] = ldsBase + (uint32_t)(2 * RPB * DEN_PITCH * 4 + r * (FEAT_PITCH * 4) + blk * 16);
  }

  // issue one chunk into buffer `bufIdx`, then advance global addrs
  auto issue = [&](uint32_t bufIdx) {
    const uint32_t bufByte = bufIdx * BUF_BYTES;   // uniform -> SALU
#pragma unroll
    for (int q = 0; q < NQ; ++q) {
      async_ld16(lO[q] + bufByte, gA[q]);
      gA[q] += (q < 4) ? (CS * 4) : (CS * 12);     // 32B or 96B per chunk
    }
  };

  // ---- pipeline prologue: chunks 0 and 1 in flight ----
  issue(0);
  issue(1);

  float T = 1.0f;
  float aF0 = 0.0f, aF1 = 0.0f, aF2 = 0.0f, aD = 0.0f;
  float pDen = 0.0f, pDep = 0.0f, pF0 = 0.0f, pF1 = 0.0f, pF2 = 0.0f;

  uint32_t sBuf = 0;        // buffer being scanned (c % 3)
  uint32_t iBuf = 2;        // buffer to issue next ((c+2) % 3)

  for (int c = 0; c < NCHUNK; ++c) {
    // chunk c's 10 loads complete once at most the next chunk's 10 remain
    if (c < NCHUNK - 1) { WAIT_ASYNCCNT(NQ); } else { WAIT_ASYNCCNT(0); }
    __syncthreads();        // single barrier per chunk (triple buffering)

    const float* base = lds + sBuf * BUF_FLOATS;
    const float* dR = base + t * DEN_PITCH;
    const float* zR = base + RPB * DEN_PITCH + t * DEN_PITCH;
    const float* fR = base + 2 * RPB * DEN_PITCH + t * FEAT_PITCH;

#pragma unroll
    for (int i = 0; i < CS; ++i) {
      const float den = dR[i];
      const float dep = zR[i];
      const float f0 = fR[3 * i + 0];
      const float f1 = fR[3 * i + 1];
      const float f2 = fR[3 * i + 2];
      if (!(c == 0 && i == 0)) {             // uniform branch; folds for i>0
        const float delta = dep - pDep;
        const float e = __expf(-pDen * delta);
        const float w = T - T * e;           // T * (1 - exp(-tau))
        aF0 = fmaf(w, pF0, aF0);
        aF1 = fmaf(w, pF1, aF1);
        aF2 = fmaf(w, pF2, aF2);
        aD  = fmaf(w, pDep, aD);
        T *= e;
      }
      pDen = den; pDep = dep; pF0 = f0; pF1 = f1; pF2 = f2;
    }

    // stage chunk c+2 into the third buffer: disjoint from scan (c) and
    // in-flight (c+1) buffers, so no trailing barrier is needed.
    if (c + 2 < NCHUNK) issue(iBuf);

    sBuf = (sBuf == NBUF - 1) ? 0u : sBuf + 1u;
    iBuf = (iBuf == NBUF - 1) ? 0u : iBuf + 1u;
  }

  // final sample absorbs remaining transmittance (delta = 1e10)
  {
    const float e = __expf(-pDen * FAR_DELTA);
    const float w = T - T * e;
    aF0 = fmaf(w, pF0, aF0);
    aF1 = fmaf(w, pF1, aF1);
    aF2 = fmaf(w, pF2, aF2);
    aD  = fmaf(w, pDep, aD);
  }

  const int ray = rayBase + t;
  if (ray < Btot) {
    out[(size_t)ray * 3 + 0] = aF0;
    out[(size_t)ray * 3 + 1] = aF1;
    out[(size_t)ray * 3 + 2] = aF2;
    out[(size_t)Btot * 3 + ray] = aD;  // depth output follows flattened feature output
  }
}

extern "C" void kernel_launch(void* const* d_in, const int* in_sizes, int n_in,
                              void* d_out, int out_size, void* d_ws, size_t ws_size,
                              hipStream_t stream) {
  const float* dens  = (const float*)d_in[0];   // (B, N, 1)
  const float* feat  = (const float*)d_in[1];   // (B, N, 3)
  const float* depth = (const float*)d_in[2];   // (B, N)
  float* out = (float*)d_out;                   // (B,3) feat then (B,1) depth, flat

  const int Btot = in_sizes[2] / NSAMP;         // 131072
  const int grid = (Btot + RPB - 1) / RPB;      // 1024
  hipLaunchKernelGGL(volrender_kernel, dim3(grid), dim3(RPB), 0, stream,
                     dens, feat, depth, out, Btot);
}